// HabitatGNN_88948772700968
// MI455X (gfx1250) — compile-verified
//
#include <hip/hip_runtime.h>
#include <hip/hip_bf16.h>

// ---------------------------------------------------------------------------
// GCN on MI455X (gfx1250). f32 end-to-end; dense X@W stages use
// V_WMMA_F32_16X16X4_F32 (wave32 matrix core, exact f32). Scatter/gather is
// the bandwidth wall and stays in L2 (192MB >> 51MB of h1+agg1).
// ---------------------------------------------------------------------------

typedef __attribute__((ext_vector_type(2))) float v2f;
typedef __attribute__((ext_vector_type(8))) float v8f;

// ---------------- degree / normalization -----------------------------------

__global__ void deg_init_kernel(float* __restrict__ deg, int n) {
    int i = blockIdx.x * blockDim.x + threadIdx.x;
    if (i < n) deg[i] = 1.0f;              // self-loop contributes 1
}

__global__ void deg_accum_kernel(const int* __restrict__ dst, float* __restrict__ deg, int e) {
    int i = blockIdx.x * blockDim.x + threadIdx.x;
    if (i < e) atomicAdd(&deg[dst[i]], 1.0f);
}

__global__ void deg_to_dinv_kernel(float* __restrict__ deg, int n) {
    int i = blockIdx.x * blockDim.x + threadIdx.x;
    if (i < n) {
        float d = deg[i];
        deg[i] = (d > 0.0f) ? rsqrtf(d) : 0.0f;   // deg >= 1 always (self-loop)
    }
}

// ---------------- WMMA f32 GEMM: C[M,N] = (reluA ? relu(A) : A) @ B ---------
// One wave computes one 16x16 tile of C. NCOLS/16 waves per block cover all
// output columns for one 16-row stripe. K stepped by 4 with
// V_WMMA_F32_16X16X4_F32. Requires M % 16 == 0 (true here: 100000/16 = 6250).
//
// VGPR layouts per CDNA5 ISA 7.12.2 (32-bit, wave32):
//   A 16x4 : lanes 0-15 rows M=0..15 hold K={k,k+1}; lanes 16-31 hold K={k+2,k+3}
//   B 4x16 : mirrored — lanes 0-15 cols N=0..15, VGPR0/1 = K rows (same half split)
//   C 16x16: VGPR v, lane L<16 -> (M = v,     N = L)
//                    lane L>=16 -> (M = v + 8, N = L-16)

template<int K, int NCOLS, bool RELU_A>
__global__ __launch_bounds__((NCOLS / 16) * 32)
void wmma_gemm_kernel(const float* __restrict__ A, const float* __restrict__ B,
                      float* __restrict__ C, int M) {
    const int wave = threadIdx.x >> 5;
    const int lane = threadIdx.x & 31;
    const int m0 = blockIdx.x * 16;
    if (m0 >= M) return;

    const int n0   = wave * 16;
    const int lo16 = lane & 15;
    const int half = lane >> 4;          // 0: lanes 0-15, 1: lanes 16-31
    const int col  = n0 + lo16;

    const float* __restrict__ arow = A + (size_t)(m0 + lo16) * K;

    v8f acc = {};
    #pragma unroll
    for (int k = 0; k < K; k += 4) {
        const int kb = k + half * 2;     // even -> 8B-aligned float2 load
        float2 av = *(const float2*)(arow + kb);
        if (RELU_A) { av.x = fmaxf(av.x, 0.0f); av.y = fmaxf(av.y, 0.0f); }
        v2f a; a[0] = av.x; a[1] = av.y;
        v2f b;
        b[0] = B[(size_t)kb * NCOLS + col];
        b[1] = B[(size_t)(kb + 1) * NCOLS + col];
        acc = __builtin_amdgcn_wmma_f32_16x16x4_f32(
                  /*neg_a=*/false, a, /*neg_b=*/false, b,
                  /*c_mod=*/(short)0, acc, /*reuse_a=*/false, /*reuse_b=*/false);
    }

    const int rbase = m0 + half * 8;
    #pragma unroll
    for (int v = 0; v < 8; ++v)
        C[(size_t)(rbase + v) * NCOLS + col] = acc[v];
}

// ---------------- aggregation: agg = bias + selfloop + scatter(edges) -------

template<int C>
__global__ void agg_init_kernel(const float* __restrict__ h, const float* __restrict__ dinv,
                                const float* __restrict__ bias, float* __restrict__ agg, int n) {
    int i = blockIdx.x * blockDim.x + threadIdx.x;
    int node = i / C, c = i % C;
    if (node < n) {
        float di = dinv[node];
        agg[i] = bias[c] + h[i] * di * di;   // self-loop norm = dinv[i]^2
    }
}

template<int C>
__global__ void edge_scatter_kernel(const int* __restrict__ src, const int* __restrict__ dst,
                                    const float* __restrict__ h, const float* __restrict__ dinv,
                                    float* __restrict__ agg, int e) {
    int t = blockIdx.x * blockDim.x + threadIdx.x;
    int edge = t / C, c = t % C;
    if (edge >= e) return;
    int s = src[edge], d = dst[edge];
    float norm = dinv[s] * dinv[d];
    atomicAdd(&agg[(size_t)d * C + c], h[(size_t)s * C + c] * norm);
}

// ---------------- head: out[i] = bc + relu(agg2[i,:]) . Wc ------------------

__global__ void head_kernel(const float* __restrict__ agg2, const float* __restrict__ Wc,
                            const float* __restrict__ bc, float* __restrict__ out, int n) {
    int i = blockIdx.x * blockDim.x + threadIdx.x;
    if (i >= n) return;
    const float* row = agg2 + (size_t)i * 32;
    float acc = bc[0];
    #pragma unroll
    for (int c = 0; c < 32; ++c)
        acc += fmaxf(row[c], 0.0f) * Wc[c];
    out[i] = acc;
}

// ---------------------------------------------------------------------------

static inline int cdiv(long long a, long long b) { return (int)((a + b - 1) / b); }

extern "C" void kernel_launch(void* const* d_in, const int* in_sizes, int n_in,
                              void* d_out, int out_size, void* d_ws, size_t ws_size,
                              hipStream_t stream) {
    const float* x   = (const float*)d_in[0];
    const int*   ei  = (const int*)d_in[1];     // [2, E] flat: src then dst
    const float* W1  = (const float*)d_in[2];
    const float* b1  = (const float*)d_in[3];
    const float* W2  = (const float*)d_in[4];
    const float* b2  = (const float*)d_in[5];
    const float* Wc  = (const float*)d_in[6];
    const float* bc  = (const float*)d_in[7];
    float* out = (float*)d_out;

    const int N = in_sizes[0] / 128;
    const int E = in_sizes[1] / 2;
    const int* src = ei;
    const int* dst = ei + E;

    // workspace layout (floats): dinv | h1[N,64] | agg1[N,64] | h2[N,32] | agg2[N,32]
    float* ws   = (float*)d_ws;
    float* dinv = ws;
    float* h1   = dinv + N;
    float* agg1 = h1 + (size_t)N * 64;
    float* h2   = agg1 + (size_t)N * 64;
    float* agg2 = h2 + (size_t)N * 32;
    (void)ws_size; (void)n_in; (void)out_size;

    const int T = 256;

    // 1) symmetric normalization dinv = rsqrt(deg + selfloop)
    deg_init_kernel<<<cdiv(N, T), T, 0, stream>>>(dinv, N);
    deg_accum_kernel<<<cdiv(E, T), T, 0, stream>>>(dst, dinv, E);
    deg_to_dinv_kernel<<<cdiv(N, T), T, 0, stream>>>(dinv, N);

    // 2) layer 1: h1 = x @ W1   (M=N, K=128, NCOLS=64) -> 4 waves / block
    wmma_gemm_kernel<128, 64, false><<<cdiv(N, 16), 128, 0, stream>>>(x, W1, h1, N);

    // 3) agg1 = b1 + selfloop + scatter
    agg_init_kernel<64><<<cdiv((long long)N * 64, T), T, 0, stream>>>(h1, dinv, b1, agg1, N);
    edge_scatter_kernel<64><<<cdiv((long long)E * 64, T), T, 0, stream>>>(src, dst, h1, dinv, agg1, E);

    // 4) layer 2: h2 = relu(agg1) @ W2   (K=64, NCOLS=32) -> 2 waves / block
    wmma_gemm_kernel<64, 32, true><<<cdiv(N, 16), 64, 0, stream>>>(agg1, W2, h2, N);

    // 5) agg2 = b2 + selfloop + scatter
    agg_init_kernel<32><<<cdiv((long long)N * 32, T), T, 0, stream>>>(h2, dinv, b2, agg2, N);
    edge_scatter_kernel<32><<<cdiv((long long)E * 32, T), T, 0, stream>>>(src, dst, h2, dinv, agg2, E);

    // 6) head: out = relu(agg2) @ Wc + bc
    head_kernel<<<cdiv(N, T), T, 0, stream>>>(agg2, Wc, bc, out, N);
}